// BloomFilterer_77661598646370
// MI455X (gfx1250) — compile-verified
//
#include <hip/hip_runtime.h>
#include <hip/hip_bf16.h>
#include <stdint.h>

// Bloom-filter negative-sample check for MI455X (gfx1250, wave32).
//
// Shape: batch (B*N, 3) int64 -> out (B*N) float (1.0 = valid / not in filter).
// The ~143.8 MB bit array fits in the 192 MB L2, so the kernel is an
// L2-gather-latency problem: 4 interleaved hash chains per thread for ILP,
// EXEC-predicated gathers with per-wave early exit, Barrett reduction instead
// of 64-bit srem, and non-temporal hints on the streaming data so it does not
// evict the L2-resident bit array.

#define BLOCK 256
#define ITEMS 4

__global__ __launch_bounds__(BLOCK) void bloom_filter_kernel(
    const long long* __restrict__ batch,      // (n_items, 3) int64
    const unsigned char* __restrict__ bits,   // (size,) bool
    const int* __restrict__ rounds_ptr,       // scalar (=10)
    float* __restrict__ out,                  // (n_items,) float
    unsigned long long size,                  // num_bits (< 2^31)
    unsigned long long magic,                 // floor((2^64 - 1) / size)
    int n_items)
{
    const int rounds = rounds_ptr[0];
    const int tile = (int)blockIdx.x * (BLOCK * ITEMS);

    int     idx[ITEMS];
    int64_t x[ITEMS];
    bool    live[ITEMS];

    #pragma unroll
    for (int k = 0; k < ITEMS; ++k) {
        idx[k]  = tile + k * BLOCK + (int)threadIdx.x;
        live[k] = (idx[k] < n_items);
        x[k]    = 0;
        if (live[k]) {
            // Streaming read-once data: non-temporal so it doesn't pollute the
            // L2 holding the bit array. 32-bit offsets (max ~3.1M elements).
            const long long* p = batch + (unsigned)(idx[k] * 3);
            uint64_t a = (uint64_t)__builtin_nontemporal_load(p + 0);
            uint64_t b = (uint64_t)__builtin_nontemporal_load(p + 1);
            uint64_t c = (uint64_t)__builtin_nontemporal_load(p + 2);
            // MERSENNE = [2^17-1, 2^19-1, 2^31-1], wraparound int64 math.
            x[k] = (int64_t)(a * 131071ull + b * 524287ull + c * 2147483647ull);
        }
    }

    for (int r = 0; r < rounds; ++r) {
        bool any = false;
        #pragma unroll
        for (int k = 0; k < ITEMS; ++k) any = any || live[k];
        if (!__any(any)) break;   // whole wave dead -> skip remaining rounds

        #pragma unroll
        for (int k = 0; k < ITEMS; ++k) {
            if (live[k]) {        // EXEC-predicated: dead lanes skip VALU+VMEM
                int64_t v = x[k];
                v ^= (v >> 16);                               // arithmetic >>
                v  = (int64_t)((uint64_t)v * 2146121005ull);  // wraparound *
                v ^= (v >> 15);
                v  = (int64_t)((uint64_t)v * 2221713035ull);
                v ^= (v >> 16);
                x[k] = v;

                // Floored modulo by `size` (NumPy semantics) via Barrett:
                // reduce |v| with magic = floor((2^64-1)/size), then fix sign.
                uint64_t a   = (v < 0) ? (0ull - (uint64_t)v) : (uint64_t)v;
                uint64_t q   = __umul64hi(a, magic);   // q in {Q-1, Q}
                uint64_t rem = a - q * size;           // size < 2^31 -> 64x32 mul
                if (rem >= size) rem -= size;
                if (rem >= size) rem -= size;
                if (v < 0 && rem != 0) rem = size - rem;

                // Bit-array gather: default (RT) temporal policy, stays in L2.
                live[k] = (bits[(unsigned)rem] != 0);
            }
        }
    }

    #pragma unroll
    for (int k = 0; k < ITEMS; ++k) {
        if (idx[k] < n_items) {
            // out = ~result: 1.0 where some probed bit was clear (valid negative)
            __builtin_nontemporal_store(live[k] ? 0.0f : 1.0f,
                                        out + (unsigned)idx[k]);
        }
    }
}

extern "C" void kernel_launch(void* const* d_in, const int* in_sizes, int n_in,
                              void* d_out, int out_size, void* d_ws, size_t ws_size,
                              hipStream_t stream) {
    // setup_inputs order:
    //   d_in[0] negative_batch (2048,512,3) int64
    //   d_in[1] bit_array      (num_bits,)  bool
    //   d_in[2] mersenne       (1,3)        int64  (module constants; hardcoded)
    //   d_in[3] rounds         scalar int
    const long long*     batch  = (const long long*)d_in[0];
    const unsigned char* bits   = (const unsigned char*)d_in[1];
    const int*           rounds = (const int*)d_in[3];
    float*               out    = (float*)d_out;

    const int n_items = in_sizes[0] / 3;
    const unsigned long long size  = (unsigned long long)in_sizes[1];
    const unsigned long long magic = ~0ull / size;   // Barrett magic, host-side

    const int per_block = BLOCK * ITEMS;
    const int grid = (n_items + per_block - 1) / per_block;

    bloom_filter_kernel<<<grid, BLOCK, 0, stream>>>(
        batch, bits, rounds, out, size, magic, n_items);
}